// GAT_R_73839077752941
// MI455X (gfx1250) — compile-verified
//
#include <hip/hip_runtime.h>

typedef __attribute__((ext_vector_type(2))) float v2f;
typedef __attribute__((ext_vector_type(4))) float v4f;
typedef __attribute__((ext_vector_type(8))) float v8f;

#define HDIM      128
#define NEG_SLOPE 0.01f
#define EPS_F     1e-16f

// ---------------------------------------------------------------------------
// Kernel 1: [s_i | s_j] = x @ [w_i | w_j] via V_WMMA_F32_16X16X4_F32.
// One wave = 16 node rows. B: col0 = w_i, col1 = w_j, other 14 cols zero.
// A layout (32-bit 16x4): lanes 0-15 = rows M (K=0,1), lanes 16-31 = rows M (K=2,3).
// Every lane loads from a VALID per-lane w pointer and scales by 0/1 — no
// divergent loads, EXEC stays all-1s (required by WMMA).
// ---------------------------------------------------------------------------
__global__ void gat_scores_wmma(const float* __restrict__ x,
                                const float* __restrict__ wi,
                                const float* __restrict__ wj,
                                float* __restrict__ s_i,
                                float* __restrict__ s_j,
                                int N) {
  const int wave = blockIdx.x * (blockDim.x >> 5) + (threadIdx.x >> 5);
  const int lane = threadIdx.x & 31;
  const int row0 = wave << 4;
  if (row0 >= N) return;                 // wave-uniform

  const int m    = lane & 15;            // A row / B column this lane feeds
  const int kofs = (lane >> 4) << 1;     // 0 (lanes 0-15) or 2 (lanes 16-31)
  int row = row0 + m;
  if (row >= N) row = N - 1;             // cndmask clamp
  const float* __restrict__ xr = x + (size_t)row * HDIM + kofs;
  const float* __restrict__ wr = ((m == 1) ? wj : wi) + kofs;  // valid for all lanes
  const float bscale = (m < 2) ? 1.0f : 0.0f;                  // zero cols 2..15

  v8f acc = {};
#pragma unroll 8
  for (int k = 0; k < HDIM; k += 4) {
    v2f a = *(const v2f*)(xr + k);       // global_load_b64
    v2f w = *(const v2f*)(wr + k);       // global_load_b64 (broadcast-ish)
    v2f b;
    b.x = bscale * w.x;
    b.y = bscale * w.y;
    acc = __builtin_amdgcn_wmma_f32_16x16x4_f32(false, a, false, b,
                                                (short)0, acc, false, false);
  }

  // D: lane = column N, VGPR v = row M=v (lanes 0-15) / M=v+8 (lanes 16-31).
  const int sel  = lane & 15;            // 0 -> s_i, 1 -> s_j
  const int half = lane >> 4;
  if (sel < 2) {
    float* __restrict__ dst = (sel == 0) ? s_i : s_j;
    const int mb = row0 + (half << 3);
#pragma unroll
    for (int v = 0; v < 8; ++v) {
      const int r = mb + v;
      if (r < N) dst[r] = acc[v];
    }
  }
}

// Monotonic float <-> uint total-order map for integer atomicMax.
__device__ __forceinline__ unsigned int f32_flip(float f) {
  int s = __float_as_int(f);
  return (unsigned int)s ^ ((s < 0) ? 0xFFFFFFFFu : 0x80000000u);
}
__device__ __forceinline__ float f32_unflip(unsigned int u) {
  int s = (int)(u ^ (((int)u < 0) ? 0x80000000u : 0xFFFFFFFFu));
  return __int_as_float(s);
}

// ---------------------------------------------------------------------------
// Kernel 2: degree histogram over destination nodes i.
// ---------------------------------------------------------------------------
__global__ void gat_degree(const int* __restrict__ ei, int* __restrict__ deg, int E) {
  const int t = blockIdx.x * blockDim.x + threadIdx.x;
  if (t >= E) return;
  const int i = __builtin_nontemporal_load(ei + E + t);
  atomicAdd(&deg[i], 1);
}

// ---------------------------------------------------------------------------
// Kernel 3: single-block exclusive scan deg -> rowptr[0..N].
// 1024 threads, Hillis-Steele per 1024-chunk with running carry.
// ---------------------------------------------------------------------------
__global__ void gat_exscan(const int* __restrict__ deg, int* __restrict__ rowptr, int N) {
  __shared__ int sdata[1024];
  __shared__ int carry;
  const int tid = threadIdx.x;
  if (tid == 0) { carry = 0; rowptr[0] = 0; }
  __syncthreads();
  for (int base = 0; base < N; base += 1024) {
    const int idx = base + tid;
    sdata[tid] = (idx < N) ? deg[idx] : 0;
    __syncthreads();
    for (int off = 1; off < 1024; off <<= 1) {
      const int v = (tid >= off) ? sdata[tid - off] : 0;
      __syncthreads();
      sdata[tid] += v;
      __syncthreads();
    }
    if (idx < N) rowptr[idx + 1] = carry + sdata[tid];   // inclusive -> [idx+1]
    __syncthreads();
    if (tid == 0) carry += sdata[1023];
    __syncthreads();
  }
}

// ---------------------------------------------------------------------------
// Kernel 4: e = leaky_relu(s_i[i] + s_j[j]); segment max over j groups.
// ---------------------------------------------------------------------------
__global__ void gat_edge_max(const int* __restrict__ ei,
                             const float* __restrict__ s_i,
                             const float* __restrict__ s_j,
                             float* __restrict__ e_buf,
                             unsigned int* __restrict__ segmax,
                             int E) {
  const int t = blockIdx.x * blockDim.x + threadIdx.x;
  if (t >= E) return;
  const int j = ei[t];
  const int i = ei[E + t];
  float e = s_i[i] + s_j[j];
  e = (e > 0.0f) ? e : NEG_SLOPE * e;
  __builtin_nontemporal_store(e, e_buf + t);
  atomicMax(&segmax[j], f32_flip(e));
}

// ---------------------------------------------------------------------------
// Kernel 5: ex = exp(e - segmax[j]); denom[j] += ex.
// ---------------------------------------------------------------------------
__global__ void gat_edge_exp(const int* __restrict__ ei,
                             const float* __restrict__ e_buf,
                             const unsigned int* __restrict__ segmax,
                             float* __restrict__ ex_buf,
                             float* __restrict__ denom,
                             int E) {
  const int t = blockIdx.x * blockDim.x + threadIdx.x;
  if (t >= E) return;
  const int j = ei[t];
  const float e = __builtin_nontemporal_load(e_buf + t);   // last use
  const float ex = __expf(e - f32_unflip(segmax[j]));
  __builtin_nontemporal_store(ex, ex_buf + t);
  atomicAdd(&denom[j], ex);
}

// ---------------------------------------------------------------------------
// Kernel 6: counting-fill CSR keyed by destination i; bake alpha in.
// ---------------------------------------------------------------------------
__global__ void gat_fill_csr(const int* __restrict__ ei,
                             const float* __restrict__ ex_buf,
                             const float* __restrict__ denom,
                             const int* __restrict__ rowptr,
                             int* __restrict__ cursor,
                             int* __restrict__ col,
                             float* __restrict__ aval,
                             int E) {
  const int t = blockIdx.x * blockDim.x + threadIdx.x;
  if (t >= E) return;
  const int j = ei[t];
  const int i = ei[E + t];
  const float alpha = __builtin_nontemporal_load(ex_buf + t) / (denom[j] + EPS_F);
  const int pos = rowptr[i] + atomicAdd(&cursor[i], 1);
  __builtin_nontemporal_store(j, col + pos);
  __builtin_nontemporal_store(alpha, aval + pos);
}

// ---------------------------------------------------------------------------
// Kernel 7: gather-SPMM + fused ReLU. One wave per node; lane owns a v4f of
// the 128-wide row. x rows are L2-resident (51 MB vs 192 MB L2); CSR arrays
// are streamed non-temporally. No atomics; each out row written exactly once.
// ---------------------------------------------------------------------------
__global__ void gat_spmm_relu(const int* __restrict__ rowptr,
                              const int* __restrict__ col,
                              const float* __restrict__ aval,
                              const float* __restrict__ x,
                              float* __restrict__ out,
                              int N) {
  const int wave = blockIdx.x * (blockDim.x >> 5) + (threadIdx.x >> 5);
  const int lane = threadIdx.x & 31;
  if (wave >= N) return;
  const int p0 = rowptr[wave];
  const int p1 = rowptr[wave + 1];
  v4f acc = {};
  for (int p = p0; p < p1; ++p) {
    const int   j = __builtin_nontemporal_load(col + p);
    const float a = __builtin_nontemporal_load(aval + p);
    const v4f v = ((const v4f*)(x + (size_t)j * HDIM))[lane];  // b128, L2 hit
    acc += a * v;                                              // v_fma x4
  }
  v4f r;
  r.x = fmaxf(acc.x, 0.0f); r.y = fmaxf(acc.y, 0.0f);
  r.z = fmaxf(acc.z, 0.0f); r.w = fmaxf(acc.w, 0.0f);
  v4f* orow = (v4f*)(out + (size_t)wave * HDIM);
  __builtin_nontemporal_store(r, orow + lane);                 // b128 NT, write-once
}

extern "C" void kernel_launch(void* const* d_in, const int* in_sizes, int n_in,
                              void* d_out, int out_size, void* d_ws, size_t ws_size,
                              hipStream_t stream) {
  const float* x  = (const float*)d_in[0];   // [N, 128]
  const int*   ei = (const int*)d_in[1];     // [2, E] : row0 = j, row1 = i
  // d_in[2] = line_graph_val : unused by the reference
  const float* wi = (const float*)d_in[3];   // [128]
  const float* wj = (const float*)d_in[4];   // [128]

  const int N = in_sizes[0] / HDIM;
  const int E = in_sizes[1] / 2;

  float* out = (float*)d_out;
  (void)out_size; (void)ws_size; (void)n_in;

  // Workspace (4B units):
  // s_i[N] | s_j[N] | segmax[N] | denom[N] | deg/cursor[N] | rowptr[N+1] |
  // e[E] | ex[E] | col[E] | aval[E]
  float*        s_i    = (float*)d_ws;
  float*        s_j    = s_i + N;
  unsigned int* segmax = (unsigned int*)(s_j + N);
  float*        denom  = (float*)(segmax + N);
  int*          deg    = (int*)(denom + N);          // reused as cursor
  int*          rowptr = deg + N;
  float*        e_buf  = (float*)(rowptr + N + 1);
  float*        ex_buf = e_buf + E;
  int*          col    = (int*)(ex_buf + E);
  float*        aval   = (float*)(col + E);

  const int eblocks = (E + 255) / 256;

  (void)hipMemsetAsync(segmax, 0, (size_t)N * sizeof(unsigned int), stream); // mapped -inf
  (void)hipMemsetAsync(denom,  0, (size_t)N * sizeof(float), stream);
  (void)hipMemsetAsync(deg,    0, (size_t)N * sizeof(int), stream);

  {  // scores via f32 WMMA: one wave per 16 rows
    const int waves  = (N + 15) / 16;
    const int blocks = (waves + 7) / 8;
    gat_scores_wmma<<<blocks, 256, 0, stream>>>(x, wi, wj, s_i, s_j, N);
  }
  gat_degree<<<eblocks, 256, 0, stream>>>(ei, deg, E);
  gat_exscan<<<1, 1024, 0, stream>>>(deg, rowptr, N);
  gat_edge_max<<<eblocks, 256, 0, stream>>>(ei, s_i, s_j, e_buf, segmax, E);
  gat_edge_exp<<<eblocks, 256, 0, stream>>>(ei, e_buf, segmax, ex_buf, denom, E);
  (void)hipMemsetAsync(deg, 0, (size_t)N * sizeof(int), stream);             // -> cursor
  gat_fill_csr<<<eblocks, 256, 0, stream>>>(ei, ex_buf, denom, rowptr, deg, col, aval, E);
  {  // gather-SPMM + ReLU: one wave per node
    const int blocks = (N + 7) / 8;
    gat_spmm_relu<<<blocks, 256, 0, stream>>>(rowptr, col, aval, x, out, N);
  }
}